// STA_Fusion_87058987090143
// MI455X (gfx1250) — compile-verified
//
#include <hip/hip_runtime.h>

typedef __attribute__((ext_vector_type(2))) float v2f;
typedef __attribute__((ext_vector_type(8))) float v8f;

#define USE_ASYNC_LDS 1   // gfx1250 GLOBAL_LOAD_ASYNC_TO_LDS path (set 0 to fall back)

namespace {
constexpr int kN = 2048, kM = 32, kD = 64, kF = 64, kB = 8, kL = 24, kH = 32;
constexpr int kRows = kB * kL;   // 192 (b,l) rows per node
constexpr int kPad  = 68;        // LDS row stride (floats): 272B -> 16B-aligned rows,
                                 // bank = (4*row + k) mod 64 -> conflict-free frags
constexpr int kThreads = 256;    // 8 wave32
}

struct Params {
  const float* G; const float* Lc;
  const float* mu[3]; const float* logvar[3]; const float* eps[3];
  const float* W1[3]; const float* b1[3];
  const float* W2[3]; const float* b2[3];
  const float* W3[3]; const float* b3[3];
  const float* projW; const float* projB; const float* lnG; const float* lnB;
  float* out;
};

__device__ __forceinline__ void async_wait0() {
#if __has_builtin(__builtin_amdgcn_s_wait_asynccnt)
  __builtin_amdgcn_s_wait_asynccnt(0);
#else
  asm volatile("s_wait_asynccnt 0" ::: "memory");
#endif
}

// One 16x16 output tile of D = A(192x64) * B(64x64) using V_WMMA_F32_16X16X4_F32.
// A-frag (16x4 f32, ISA 7.12.2): lanes 0-15 = rows M0..15 with K={0,1} in v0/v1,
// lanes 16-31 = same rows with K={2,3}. B-frag mirrors. D: VGPR r -> M=r / r+8.
__device__ __forceinline__ void wmma_tile_f32(
    const float* __restrict__ sA,   // [192][kPad]
    const float* __restrict__ sB,   // [64][kPad]
    float*       __restrict__ sD,   // [192][kPad]
    int mt, int nt, int lane)
{
  const int lr = lane & 15;
  const int lh = lane >> 4;
  v8f acc = {};
  const float* aRow = sA + (mt * 16 + lr) * kPad;
  const float* bCol = sB + nt * 16 + lr;
#pragma unroll
  for (int kk = 0; kk < 16; ++kk) {
    const int kb = kk * 4 + lh * 2;
    v2f a, b;
    a.x = aRow[kb];
    a.y = aRow[kb + 1];
    b.x = bCol[kb * kPad];
    b.y = bCol[(kb + 1) * kPad];
    acc = __builtin_amdgcn_wmma_f32_16x16x4_f32(false, a, false, b,
                                                (short)0, acc, false, false);
  }
  float* dBase = sD + (mt * 16 + lh * 8) * kPad + nt * 16 + lr;
#pragma unroll
  for (int r = 0; r < 8; ++r) dBase[r * kPad] = acc[r];
}

__global__ __launch_bounds__(kThreads)
void sta_fusion_wmma(Params p)
{
  extern __shared__ float sm[];
  float* sG  = sm;                   // global_embed rows      [192][68]
  float* sLQ = sG  + kRows * kPad;   // local_embed -> then Q  [192][68]
  float* sK  = sLQ + kRows * kPad;   // K -> then proj P       [192][68]
  float* sV  = sK  + kRows * kPad;   // V                      [192][68]
  float* sW  = sV  + kRows * kPad;   // Wq|Wk|Wv -> then attnO [3*64][68]
  float* sPW = sW  + 3 * 64 * kPad;  // proj_W                 [64][68]
  float* sSm = sPW + 64 * kPad;
  float* sZ  = sSm;                  // 96
  float* sH1 = sZ  + 96;             // 96
  float* sH2 = sH1 + 96;             // 96
  float* sPB = sH2 + 96;             // 64
  float* sLG = sPB + 64;             // 64
  float* sLB = sLG + 64;             // 64

  const int n    = blockIdx.x;
  const int t    = threadIdx.x;
  const int lane = t & 31;
  const int wave = t >> 5;

  // ---- embeddings: 2 x 48KB, 16B/lane straight into LDS (ASYNCcnt path) ----
#if USE_ASYNC_LDS
  for (int i = t; i < (kB * kL * kD) / 4; i += kThreads) {
    const int b    = i / (kL * kD / 4);        // 384 vec4 per batch
    const int remv = i - b * (kL * kD / 4);
    const int row  = b * kL + (remv >> 4);     // 16 vec4 per row
    const int dv   = (remv & 15) * 4;
    const size_t gi = ((size_t)(b * kN + n) * kL) * kD + (size_t)remv * 4;
    const unsigned gOff = (unsigned)(uintptr_t)(sG  + row * kPad + dv);
    const unsigned lOff = (unsigned)(uintptr_t)(sLQ + row * kPad + dv);
    asm volatile("global_load_async_to_lds_b128 %0, %1, off"
                 :: "v"(gOff), "v"((unsigned long long)(uintptr_t)(p.G + gi))
                 : "memory");
    asm volatile("global_load_async_to_lds_b128 %0, %1, off"
                 :: "v"(lOff), "v"((unsigned long long)(uintptr_t)(p.Lc + gi))
                 : "memory");
  }
#else
  for (int i = t; i < (kB * kL * kD) / 4; i += kThreads) {
    const int b    = i / (kL * kD / 4);
    const int remv = i - b * (kL * kD / 4);
    const int row  = b * kL + (remv >> 4);
    const int dv   = (remv & 15) * 4;
    const size_t gi = ((size_t)(b * kN + n) * kL) * kD + (size_t)remv * 4;
    *(float4*)(sG  + row * kPad + dv) = *(const float4*)(p.G  + gi);
    *(float4*)(sLQ + row * kPad + dv) = *(const float4*)(p.Lc + gi);
  }
#endif
  // ---- proj_W (16B vectors), LN params, reparameterized latents ----
  for (int i = t; i < 64 * 64 / 4; i += kThreads) {
    const int r = i >> 4, c = (i & 15) * 4;
    *(float4*)(sPW + r * kPad + c) = *(const float4*)(p.projW + (size_t)i * 4);
  }
  if (t < 64) { sPB[t] = p.projB[t]; sLG[t] = p.lnG[t]; sLB[t] = p.lnB[t]; }
  if (t < 96) {
    const int pp = t / 32, m = t & 31;
    const float lv = p.logvar[pp][n * kM + m];
    sZ[t] = p.mu[pp][n * kM + m] + __expf(0.5f * lv) * p.eps[pp][n * kM + m];
  }
#if USE_ASYNC_LDS
  async_wait0();
#endif
  __syncthreads();

  // ---- hypernetwork MLP (tiny; W1/W2/W3 stay hot in L2) ----
  if (t < 96) {
    const int pp = t / 32, j = t & 31;
    float acc = p.b1[pp][j];
#pragma unroll
    for (int m = 0; m < kM; ++m) acc += sZ[pp * 32 + m] * p.W1[pp][m * kH + j];
    sH1[t] = fmaxf(acc, 0.0f);
  }
  __syncthreads();
  if (t < 96) {
    const int pp = t / 32, j = t & 31;
    float acc = p.b2[pp][j];
#pragma unroll
    for (int m = 0; m < kH; ++m) acc += sH1[pp * 32 + m] * p.W2[pp][m * kH + j];
    sH2[t] = fmaxf(acc, 0.0f);
  }
  __syncthreads();
  // W_p = h2 @ W3_p + b3_p  -> 3 x (64x64) generated weights in LDS
  for (int o = t; o < 3 * kD * kF; o += kThreads) {
    const int pp = o >> 12;
    const int df = o & 4095;
    float acc = p.b3[pp][df];
    const float* w3 = p.W3[pp] + df;
    const float* h2 = sH2 + pp * 32;
#pragma unroll
    for (int j = 0; j < kH; ++j) acc += h2[j] * w3[j * (kD * kF)];
    sW[pp * (64 * kPad) + (df >> 6) * kPad + (df & 63)] = acc;
  }
  __syncthreads();

  // ---- K = local @ Wk ; V = local @ Wv  (96 WMMA tiles over 8 waves) ----
  for (int tt = wave; tt < 96; tt += 8) {
    const int g = tt / 48, tile = tt - g * 48;
    wmma_tile_f32(sLQ, sW + (1 + g) * 64 * kPad, g ? sV : sK,
                  tile >> 2, tile & 3, lane);
  }
  __syncthreads();
  // ---- Q = global @ Wq (overwrite local buffer) ----
  for (int tt = wave; tt < 48; tt += 8)
    wmma_tile_f32(sG, sW, sLQ, tt >> 2, tt & 3, lane);
  __syncthreads();

  // ---- attention: softmax(Q K^T / 8) V, one thread per (b,l) row ----
  if (t < kRows) {
    const int b = t / kL;
    const float4* q4 = (const float4*)(sLQ + t * kPad);
    float4 qr[16];
#pragma unroll
    for (int d = 0; d < 16; ++d) qr[d] = q4[d];
    float s[kL];
    float mx = -3.0e38f;
#pragma unroll
    for (int k = 0; k < kL; ++k) {
      const float4* kr = (const float4*)(sK + (b * kL + k) * kPad);
      float4 a4 = {0.f, 0.f, 0.f, 0.f};
#pragma unroll
      for (int d = 0; d < 16; ++d) {
        const float4 kv = kr[d];
        a4.x += qr[d].x * kv.x; a4.y += qr[d].y * kv.y;
        a4.z += qr[d].z * kv.z; a4.w += qr[d].w * kv.w;
      }
      const float acc = ((a4.x + a4.y) + (a4.z + a4.w)) * 0.125f; // 1/sqrt(F)
      s[k] = acc;
      mx = fmaxf(mx, acc);
    }
    float den = 0.0f;
#pragma unroll
    for (int k = 0; k < kL; ++k) { s[k] = __expf(s[k] - mx); den += s[k]; }
    const float inv = 1.0f / den;
    float* orow = sW + t * kPad;           // reuse generated-weight space
    for (int f = 0; f < kF; f += 4) {
      float4 a4 = {0.f, 0.f, 0.f, 0.f};
#pragma unroll
      for (int k = 0; k < kL; ++k) {
        const float4 v = *(const float4*)(sV + (b * kL + k) * kPad + f);
        a4.x += s[k] * v.x; a4.y += s[k] * v.y;
        a4.z += s[k] * v.z; a4.w += s[k] * v.w;
      }
      float4 o4; o4.x = a4.x * inv; o4.y = a4.y * inv;
      o4.z = a4.z * inv; o4.w = a4.w * inv;
      *(float4*)(orow + f) = o4;
    }
  }
  __syncthreads();

  // ---- residual projection P = global @ proj_W (reuse K buffer) ----
  for (int tt = wave; tt < 48; tt += 8)
    wmma_tile_f32(sG, sPW, sK, tt >> 2, tt & 3, lane);
  __syncthreads();

  // ---- LayerNorm over F=64 + store, one wave per row (b64 stores) ----
  for (int r = wave; r < kRows; r += 8) {
    const int f0 = 2 * lane;
    const float2 o2 = *(const float2*)(sW + r * kPad + f0);
    const float2 q2 = *(const float2*)(sK + r * kPad + f0);
    const float x0 = o2.x + q2.x + sPB[f0];
    const float x1 = o2.y + q2.y + sPB[f0 + 1];
    float sum = x0 + x1, sq = x0 * x0 + x1 * x1;
#pragma unroll
    for (int off = 16; off; off >>= 1) {
      sum += __shfl_xor(sum, off, 32);
      sq  += __shfl_xor(sq,  off, 32);
    }
    const float mean = sum * (1.0f / 64.0f);
    const float var  = sq * (1.0f / 64.0f) - mean * mean;
    const float rstd = rsqrtf(var + 1e-5f);
    const int b = r / kL, l = r - b * kL;
    const size_t oi = (((size_t)(b * kN + n)) * kL + l) * (size_t)kF;
    float2 y;
    y.x = (x0 - mean) * rstd * sLG[f0]     + sLB[f0];
    y.y = (x1 - mean) * rstd * sLG[f0 + 1] + sLB[f0 + 1];
    *(float2*)(p.out + oi + f0) = y;
  }
}

extern "C" void kernel_launch(void* const* d_in, const int* in_sizes, int n_in,
                              void* d_out, int out_size, void* d_ws, size_t ws_size,
                              hipStream_t stream) {
  (void)in_sizes; (void)n_in; (void)out_size; (void)d_ws; (void)ws_size;
  Params p;
  p.G  = (const float*)d_in[0];
  p.Lc = (const float*)d_in[1];
  p.mu[0]     = (const float*)d_in[2];
  p.logvar[0] = (const float*)d_in[3];
  p.mu[1]     = (const float*)d_in[4];
  p.logvar[1] = (const float*)d_in[5];
  p.mu[2]     = (const float*)d_in[6];
  p.logvar[2] = (const float*)d_in[7];
  p.eps[0] = (const float*)d_in[8];
  p.eps[1] = (const float*)d_in[9];
  p.eps[2] = (const float*)d_in[10];
  for (int j = 0; j < 3; ++j) {
    const int base = 11 + j * 6;
    p.W1[j] = (const float*)d_in[base + 0];
    p.b1[j] = (const float*)d_in[base + 1];
    p.W2[j] = (const float*)d_in[base + 2];
    p.b2[j] = (const float*)d_in[base + 3];
    p.W3[j] = (const float*)d_in[base + 4];
    p.b3[j] = (const float*)d_in[base + 5];
  }
  p.projW = (const float*)d_in[29];
  p.projB = (const float*)d_in[30];
  p.lnG   = (const float*)d_in[31];
  p.lnB   = (const float*)d_in[32];
  p.out   = (float*)d_out;

  const size_t smem =
      (size_t)(5 * kRows * kPad + 64 * kPad + 512) * sizeof(float); // ~280 KB (< 320 KB WGP LDS)
  (void)hipFuncSetAttribute(reinterpret_cast<const void*>(sta_fusion_wmma),
                            hipFuncAttributeMaxDynamicSharedMemorySize, (int)smem);
  sta_fusion_wmma<<<kN, kThreads, smem, stream>>>(p);
}